// GATClassifier_37924561224135
// MI455X (gfx1250) — compile-verified
//
#include <hip/hip_runtime.h>
#include <hip/hip_bf16.h>
#include <math.h>

#define IN_CH 128
#define HID   64
#define NEG_SLOPE 0.2f
#define MAX_K 128
#define MAX_NC 64

typedef __attribute__((ext_vector_type(2))) float v2f;
typedef __attribute__((ext_vector_type(8))) float v8f;

// ---------------------------------------------------------------------------
// Small transpose: in (rows x cols) -> out (cols x rows)
// ---------------------------------------------------------------------------
__global__ void transpose_kernel(const float* __restrict__ in, float* __restrict__ out,
                                 int rows, int cols)
{
    int i = blockIdx.x * blockDim.x + threadIdx.x;
    if (i >= rows * cols) return;
    int r = i / cols, c = i % cols;
    out[c * rows + r] = in[i];
}

// ---------------------------------------------------------------------------
// FP32 WMMA GEMM with LDS-staged B:
//   D[m][dcol0+n] = sum_k A[m][k]*B[k][n] (+ bias[n])
// B is K x Ncols row-major, contiguous (ldb == Ncols), K*Ncols <= 128*64.
// Whole B staged once per block into padded LDS (stride Nc+1 -> no bank
// conflicts between half-waves), then one wave32 per 16x16 output tile,
// K stepped by 4 via v_wmma_f32_16x16x4_f32.
// A-layout (16x4 f32): lane L holds row M=L%16, VGPR0=K=kh, VGPR1=K=kh+1, kh=2*(L/16)
// B-layout (4x16 f32): lane L holds col N=L%16 of rows {kh, kh+1}
// D-layout (16x16 f32): VGPR r, lane L -> row r + 8*(L/16), col L%16
// ---------------------------------------------------------------------------
__global__ __launch_bounds__(256)
void wmma_gemm_f32(const float* __restrict__ A, int lda,
                   const float* __restrict__ B,
                   const float* __restrict__ bias,
                   float* __restrict__ D, int ldd, int dcol0,
                   int M, int K, int Ncols)
{
    __shared__ float Bs[MAX_K * (MAX_NC + 1)];
    const int ldsb = Ncols + 1;

    // cooperative stage of B (float4 global reads, scalar LDS writes)
    const int tot4 = (K * Ncols) >> 2;
    for (int i4 = threadIdx.x; i4 < tot4; i4 += blockDim.x) {
        const int i = i4 << 2;
        const float4 v = *reinterpret_cast<const float4*>(B + i);
        const int r = i / Ncols, c = i % Ncols;   // Ncols % 4 == 0 -> no row crossing
        float* p = Bs + r * ldsb + c;
        p[0] = v.x; p[1] = v.y; p[2] = v.z; p[3] = v.w;
    }
    __syncthreads();

    const int lane = threadIdx.x & 31;
    const int wave = (blockIdx.x * blockDim.x + threadIdx.x) >> 5;
    const int tilesPerRow = Ncols >> 4;
    const int nTiles = (M >> 4) * tilesPerRow;
    if (wave >= nTiles) return;
    const int row0 = (wave / tilesPerRow) << 4;
    const int col0 = (wave % tilesPerRow) << 4;
    const int ml = lane & 15;          // A-row / B-col / D-col within tile
    const int kh = (lane >> 4) << 1;   // 0 (lanes 0-15) or 2 (lanes 16-31)

    v8f acc = {};
    const float* ap = A + (size_t)(row0 + ml) * lda + kh;
    const float* bs = Bs + col0 + ml;
    for (int k = 0; k < K; k += 4) {
        // WGP-scope prefetch (locality 3 -> nearest scope: pulls into all
        // cache levels, incl. WGP$) one chunk ahead on the A stream
        __builtin_prefetch(ap + 16, 0, 3);
        v2f a, b;
        a.x = ap[0];
        a.y = ap[1];
        b.x = bs[(k + kh) * ldsb];
        b.y = bs[(k + kh + 1) * ldsb];
        acc = __builtin_amdgcn_wmma_f32_16x16x4_f32(
            /*neg_a=*/false, a, /*neg_b=*/false, b,
            /*c_mod=*/(short)0, acc, /*reuse_a=*/false, /*reuse_b=*/false);
        ap += 4;
    }

    const float bv = bias ? bias[col0 + ml] : 0.0f;
    const int rbase = row0 + ((lane >> 4) << 3);
    float* dp = D + dcol0 + col0 + ml;
#pragma unroll
    for (int r = 0; r < 8; ++r)
        dp[(size_t)(rbase + r) * ldd] = acc[r] + bv;
}

// ---------------------------------------------------------------------------
// Attention score dots: s_i[n,h] = <xh[n,h,:], a_i[h]>, s_j likewise
// xh layout: (n*H + h)*C + c ; a layout: a[h*2C + c] (a_i) / a[h*2C + C + c] (a_j)
// ---------------------------------------------------------------------------
__global__ void scores_kernel(const float* __restrict__ xh, const float* __restrict__ a,
                              float* __restrict__ s_i, float* __restrict__ s_j,
                              int Nn, int H, int C)
{
    int idx = blockIdx.x * blockDim.x + threadIdx.x;
    if (idx >= Nn * H) return;
    int h = idx % H;
    const float* xr = xh + (size_t)idx * C;
    const float* ai = a + (size_t)h * 2 * C;
    const float* aj = ai + C;
    float si = 0.f, sj = 0.f;
    for (int c = 0; c < C; ++c) {
        float v = xr[c];
        si = fmaf(v, ai[c], si);
        sj = fmaf(v, aj[c], sj);
    }
    s_i[idx] = si;
    s_j[idx] = sj;
}

// ---------------------------------------------------------------------------
// Init mutable state every launch (graph-replay determinism)
// ---------------------------------------------------------------------------
__global__ void init_kernel(unsigned* __restrict__ mkey, float* __restrict__ denom,
                            float* __restrict__ outacc, int nh, int nout)
{
    int i = blockIdx.x * blockDim.x + threadIdx.x;
    if (i < nh) { mkey[i] = 0u; denom[i] = 0.f; }
    if (i < nout) outacc[i] = 0.f;
}

// ---------------------------------------------------------------------------
// Edge helpers
// ---------------------------------------------------------------------------
__device__ __forceinline__ void edge_nodes(const long long* __restrict__ ei,
                                           int e, int E0v, int& s, int& d)
{
    if (e < E0v) { s = (int)ei[e]; d = (int)ei[E0v + e]; }
    else         { s = e - E0v;    d = s; }
}

__device__ __forceinline__ float leakyr(float v) { return v > 0.f ? v : NEG_SLOPE * v; }

// order-preserving float <-> uint key for atomicMax over signed floats
__device__ __forceinline__ unsigned fkey(float f)
{
    unsigned u = __float_as_uint(f);
    return (u & 0x80000000u) ? ~u : (u ^ 0x80000000u);
}
__device__ __forceinline__ float kfloat(unsigned k)
{
    unsigned u = (k & 0x80000000u) ? (k ^ 0x80000000u) : ~k;
    return __uint_as_float(u);
}

// pass 1: alpha = leaky(s_i[dst]+s_j[src]); segment max via ordered-uint keys;
//         cache alpha per (edge, head)
__global__ void edge_max_kernel(const long long* __restrict__ ei,
                                const float* __restrict__ s_i, const float* __restrict__ s_j,
                                unsigned* __restrict__ mkey, float* __restrict__ ebuf,
                                int E0v, int Etot, int H)
{
    int idx = blockIdx.x * blockDim.x + threadIdx.x;
    if (idx >= Etot * H) return;
    int e = idx / H, h = idx % H;
    int s, d;
    edge_nodes(ei, e, E0v, s, d);
    float alpha = leakyr(s_i[d * H + h] + s_j[s * H + h]);
    ebuf[idx] = alpha;
    atomicMax(&mkey[d * H + h], fkey(alpha));
}

// pass 2: ev = exp(alpha - max); denom[dst,h] += ev; cache ev
__global__ void edge_sum_kernel(const long long* __restrict__ ei,
                                const unsigned* __restrict__ mkey, float* __restrict__ denom,
                                float* __restrict__ ebuf,
                                int E0v, int Etot, int H)
{
    int idx = blockIdx.x * blockDim.x + threadIdx.x;
    if (idx >= Etot * H) return;
    int e = idx / H, h = idx % H;
    int s, d;
    edge_nodes(ei, e, E0v, s, d);
    float ev = expf(ebuf[idx] - kfloat(mkey[d * H + h]));
    ebuf[idx] = ev;
    atomicAdd(&denom[d * H + h], ev);
}

// pass 3: out[dst,h,:] += xh[src,h,:] * ev/(denom+eps)
// one thread per (edge, head, 16-channel group); float4 gathers, scalar atomics
__global__ void edge_aggr_kernel(const long long* __restrict__ ei,
                                 const float* __restrict__ ebuf, const float* __restrict__ denom,
                                 const float* __restrict__ xh, float* __restrict__ outacc,
                                 int E0v, int Etot, int H, int C)
{
    const int CG = C >> 4;
    int idx = blockIdx.x * blockDim.x + threadIdx.x;
    if (idx >= Etot * H * CG) return;
    int cg = idx % CG;
    int t = idx / CG;
    int h = t % H;
    int e = t / H;
    int s, d;
    edge_nodes(ei, e, E0v, s, d);
    float w = ebuf[e * H + h] / (denom[d * H + h] + 1e-16f);
    const float4* xr =
        reinterpret_cast<const float4*>(xh + ((size_t)s * H + h) * C) + cg * 4;
    float* orow = outacc + ((size_t)d * H + h) * C + cg * 16;
#pragma unroll
    for (int q = 0; q < 4; ++q) {
        float4 v = xr[q];
        atomicAdd(&orow[q * 4 + 0], v.x * w);
        atomicAdd(&orow[q * 4 + 1], v.y * w);
        atomicAdd(&orow[q * 4 + 2], v.z * w);
        atomicAdd(&orow[q * 4 + 3], v.w * w);
    }
}

__global__ void elu_kernel(const float* __restrict__ in, float* __restrict__ out, int n)
{
    int i = blockIdx.x * blockDim.x + threadIdx.x;
    if (i >= n) return;
    float v = in[i];
    out[i] = v > 0.f ? v : (expf(v) - 1.f);
}

// ---------------------------------------------------------------------------
// Host orchestration
// ---------------------------------------------------------------------------
extern "C" void kernel_launch(void* const* d_in, const int* in_sizes, int n_in,
                              void* d_out, int out_size, void* d_ws, size_t ws_size,
                              hipStream_t stream)
{
    const float*     x     = (const float*)d_in[0];
    const long long* ei    = (const long long*)d_in[1];  // int64 edge_index (2 x E0)
    const float*     emb_w = (const float*)d_in[2];      // (64,128)
    const float*     emb_b = (const float*)d_in[3];      // (64,)
    const float*     w0    = (const float*)d_in[4];      // (2,64,64)
    const float*     a0    = (const float*)d_in[5];      // (1,2,128)
    const float*     w1    = (const float*)d_in[6];      // (2,128,64)
    const float*     a1    = (const float*)d_in[7];      // (1,2,128)
    const float*     w2    = (const float*)d_in[8];      // (1,128,64)
    const float*     a2    = (const float*)d_in[9];      // (1,1,128)
    float* out = (float*)d_out;

    const int Nn   = in_sizes[0] / IN_CH;  // 50000 (multiple of 16)
    const int E0v  = in_sizes[1] / 2;      // 800000
    const int Etot = E0v + Nn;             // + self loops

    // carve scratch from d_ws
    char* ws = (char*)d_ws;
    size_t off = 0;
    auto carve = [&](size_t bytes) -> void* {
        void* p = ws + off;
        off += (bytes + 255) & ~(size_t)255;
        return p;
    };
    float*    bufA   = (float*)carve((size_t)Nn * 128 * sizeof(float));
    float*    bufXH  = (float*)carve((size_t)Nn * 128 * sizeof(float));
    float*    bufOut = (float*)carve((size_t)Nn * 128 * sizeof(float));
    float*    ebuf   = (float*)carve((size_t)Etot * 2 * sizeof(float));
    float*    s_i    = (float*)carve((size_t)Nn * 2 * sizeof(float));
    float*    s_j    = (float*)carve((size_t)Nn * 2 * sizeof(float));
    unsigned* mkey   = (unsigned*)carve((size_t)Nn * 2 * sizeof(unsigned));
    float*    denom  = (float*)carve((size_t)Nn * 2 * sizeof(float));
    float*    embT   = (float*)carve((size_t)IN_CH * HID * sizeof(float));

    const int TB = 256;
    auto nblk = [](long long n, int tb) { return (int)((n + tb - 1) / tb); };

    // emb_w (64x128) -> embT (128x64)
    transpose_kernel<<<nblk((long long)HID * IN_CH, TB), TB, 0, stream>>>(emb_w, embT, HID, IN_CH);

    auto gemm = [&](const float* A, int lda, const float* B, const float* bias,
                    float* D, int ldd, int dcol0, int M, int K, int Nc) {
        int tiles = (M / 16) * (Nc / 16);
        int wavesPerBlock = TB / 32;  // wave32
        wmma_gemm_f32<<<nblk(tiles, wavesPerBlock), TB, 0, stream>>>(
            A, lda, B, bias, D, ldd, dcol0, M, K, Nc);
    };

    // embedding: bufA[N,64] = x @ emb_w^T + emb_b
    gemm(x, IN_CH, embT, emb_b, bufA, HID, 0, Nn, IN_CH, HID);

    auto gat = [&](const float* in, int inF, const float* w, const float* a, int H,
                   float* result /* N x H*C, ELU applied */) {
        const int C = HID;  // 64 for every layer (OUT_CH == HID)
        // xh[n,h,:] = in[n,:] @ w[h]   (w[h] is inF x C row-major)
        for (int h = 0; h < H; ++h)
            gemm(in, inF, w + (size_t)h * inF * C, nullptr, bufXH, H * C, h * C, Nn, inF, C);

        long long nh = (long long)Nn * H, nout = (long long)Nn * H * C;
        init_kernel<<<nblk(nout, TB), TB, 0, stream>>>(mkey, denom, bufOut, (int)nh, (int)nout);
        scores_kernel<<<nblk(nh, TB), TB, 0, stream>>>(bufXH, a, s_i, s_j, Nn, H, C);
        edge_max_kernel<<<nblk((long long)Etot * H, TB), TB, 0, stream>>>(
            ei, s_i, s_j, mkey, ebuf, E0v, Etot, H);
        edge_sum_kernel<<<nblk((long long)Etot * H, TB), TB, 0, stream>>>(
            ei, mkey, denom, ebuf, E0v, Etot, H);
        edge_aggr_kernel<<<nblk((long long)Etot * H * (C / 16), TB), TB, 0, stream>>>(
            ei, ebuf, denom, bufXH, bufOut, E0v, Etot, H, C);
        elu_kernel<<<nblk(nout, TB), TB, 0, stream>>>(bufOut, result, (int)nout);
    };

    gat(bufA, HID, w0, a0, 2, bufA);  // layer0: N x 64  -> N x 128 (gemm reads bufA before elu writes it)
    gat(bufA, 128, w1, a1, 2, bufA);  // layer1: N x 128 -> N x 128
    gat(bufA, 128, w2, a2, 1, out);   // layer2: N x 128 -> N x 64, mean over 1 head == identity, into d_out
}